// LiLTRobertaLikeForRelationExtraction_15977278341602
// MI455X (gfx1250) — compile-verified
//
#include <hip/hip_runtime.h>
#include <hip/hip_bf16.h>

typedef __attribute__((ext_vector_type(2))) float v2f;
typedef __attribute__((ext_vector_type(8))) float v8f;

#define BS 2
#define L  1024
#define Hd 768
#define C  16
#define ROWS (BS * L)          // 2048
#define W_STRIDE (2 * Hd)      // 1536

// ---------------------------------------------------------------------------
// Kernel 1: Pi = hs · W1^T ; Pj = hs · W2^T + bias   (both ROWS x 16, fp32)
// One wave32 per 16-row tile of hs; V_WMMA_F32_16X16X4_F32, K-loop of 192 steps,
// two WMMAs (W1 half / W2 half) per step. N=16 covers all C outputs at once.
// ---------------------------------------------------------------------------
__global__ __launch_bounds__(256) void proj_wmma_kernel(
    const float* __restrict__ hs,   // (ROWS, 768)
    const float* __restrict__ W,    // (16, 1536)
    const float* __restrict__ bias, // (16)
    float* __restrict__ Pi,         // (ROWS, 16)
    float* __restrict__ Pj)         // (ROWS, 16)
{
    const int lane = threadIdx.x & 31;
    const int wave = threadIdx.x >> 5;           // 0..7
    const int tile = blockIdx.x * 8 + wave;      // 0..127 (ROWS/16 tiles)

    const int mn   = lane & 15;                  // row of A frag / col of B frag
    const int half = lane >> 4;                  // 0 or 1 -> K sub-offset 0 or 2

    const float* arow  = hs + (size_t)(tile * 16 + mn) * Hd;   // A[m][*]
    const float* w1row = W  + (size_t)mn * W_STRIDE;           // B1[k][n] = W[n][k]
    const float* w2row = w1row + Hd;                           // B2[k][n] = W[n][H+k]

    v8f acc1 = {};
    v8f acc2 = {};

    #pragma unroll 4
    for (int k0 = 0; k0 < Hd; k0 += 4) {
        const int k = k0 + half * 2;
        v2f a  = *(const v2f*)(arow  + k);   // A[m][k], A[m][k+1]
        v2f b1 = *(const v2f*)(w1row + k);   // B1[k][n], B1[k+1][n]
        v2f b2 = *(const v2f*)(w2row + k);
        acc1 = __builtin_amdgcn_wmma_f32_16x16x4_f32(
            false, a, false, b1, (short)0, acc1, false, false);
        acc2 = __builtin_amdgcn_wmma_f32_16x16x4_f32(
            false, a, false, b2, (short)0, acc2, false, false);
    }

    // D layout: VGPR v holds row (v + 8*half), col mn.
    const float bn = bias[mn];
    #pragma unroll
    for (int v = 0; v < 8; ++v) {
        const int r = tile * 16 + v + 8 * half;
        Pi[(size_t)r * C + mn] = acc1[v];
        Pj[(size_t)r * C + mn] = acc2[v] + bn;   // fold bias into Pj
    }
}

// ---------------------------------------------------------------------------
// Kernel 2: out[b,i,j,c] = Pi[b,i,c] + Pj[b,j,c]  -- the 128 MB store stream.
// One block per (b,i) row: Pi row (16 floats) held in registers as a float4,
// Pj (128 KB total) is L2-resident. float4 stores, consecutive lanes write
// consecutive float4 -> fully coalesced 512 B per wave-iteration.
// ---------------------------------------------------------------------------
__global__ __launch_bounds__(256) void bcast_add_kernel(
    const float4* __restrict__ Pi,   // (ROWS, 4) float4
    const float4* __restrict__ Pj,   // (ROWS, 4) float4
    float4* __restrict__ out)        // (ROWS, L, 4) float4
{
    const int bi = blockIdx.x;          // b*L + i, 0..2047
    const int b  = bi >> 10;
    const int t  = threadIdx.x;         // 0..255
    const int c4 = t & 3;               // which float4 of the 16 channels

    const float4 pi = Pi[(size_t)bi * 4 + c4];
    const float4* __restrict__ pjb = Pj + (size_t)b * L * 4;
    float4* __restrict__ ob        = out + (size_t)bi * L * 4;

    #pragma unroll 4
    for (int j = t >> 2; j < L; j += 64) {
        const float4 pj = pjb[(size_t)j * 4 + c4];
        float4 r;
        r.x = pi.x + pj.x;
        r.y = pi.y + pj.y;
        r.z = pi.z + pj.z;
        r.w = pi.w + pj.w;
        ob[(size_t)j * 4 + c4] = r;
    }
}

extern "C" void kernel_launch(void* const* d_in, const int* in_sizes, int n_in,
                              void* d_out, int out_size, void* d_ws, size_t ws_size,
                              hipStream_t stream) {
    const float* hs   = (const float*)d_in[0];   // (2,1024,768) f32
    const float* W    = (const float*)d_in[1];   // (16,1536) f32
    const float* bias = (const float*)d_in[2];   // (16) f32
    float* out = (float*)d_out;                  // (2,1024,1024,16) f32

    float* Pi = (float*)d_ws;                    // ROWS*16 floats = 128 KB
    float* Pj = Pi + (size_t)ROWS * C;           // next 128 KB

    // 128 tiles of 16 rows, 8 waves per block -> 16 blocks
    proj_wmma_kernel<<<dim3(ROWS / 16 / 8), dim3(256), 0, stream>>>(hs, W, bias, Pi, Pj);

    // one block per (b,i) row -> 2048 blocks, each writes 64 KB
    bcast_add_kernel<<<dim3(ROWS), dim3(256), 0, stream>>>(
        (const float4*)Pi, (const float4*)Pj, (float4*)out);
}